// Model_47536698032545
// MI455X (gfx1250) — compile-verified
//
#include <hip/hip_runtime.h>
#include <math.h>

// ---------------------------------------------------------------------------
// MI455X (gfx1250) implementation.
// Dominant compute: fp32 complex per-mode GEMMs + forward-DFT GEMMs, all done
// with V_WMMA_F32_16X16X4_F32 (native CDNA5 fp32 WMMA). Everything else is
// bandwidth-light VALU glue. The reference only consumes m[:, DEC_LEN-1], so
// the irFFT/upsample tail is reduced to single-column inverse DFTs and a
// 3-step W_up_1 chain (mathematically identical to the reference).
// ---------------------------------------------------------------------------

typedef float v2f __attribute__((ext_vector_type(2)));
typedef float v8f __attribute__((ext_vector_type(8)));

#define B_N    32
#define C_N    64
#define SEQ_N  2051
#define PRED_N 515
#define BC_N   2048
#define D_N    32
#define CK_N   128
#define NENC   256
#define MROWS  (BC_N * CK_N)   /* 262144 GEMM rows for (bc,i) flattening */

static __device__ __forceinline__ v8f v8zero() {
  v8f z = {0.f, 0.f, 0.f, 0.f, 0.f, 0.f, 0.f, 0.f};
  return z;
}

static __device__ __forceinline__ v8f wmma_f32_16x16x4(v2f a, v2f b, v8f c) {
  // 8-arg pattern: (neg_a, A, neg_b, B, c_mod, C, reuse_a, reuse_b)
  return __builtin_amdgcn_wmma_f32_16x16x4_f32(false, a, false, b, (short)0, c,
                                               false, false);
}

// ---------------------------------------------------------------------------
// 1) per-(b,c) mean / stdev over SEQ
// ---------------------------------------------------------------------------
__global__ void stats_kernel(const float* __restrict__ x, float* __restrict__ mean,
                             float* __restrict__ stdev, float* __restrict__ rstd) {
  const int bc = blockIdx.x;
  const int b = bc >> 6, c = bc & 63;
  const float* p = x + (size_t)b * SEQ_N * C_N + c;
  float s = 0.f, s2 = 0.f;
  for (int t = threadIdx.x; t < SEQ_N; t += 256) {
    float v = p[(size_t)t * C_N];
    s += v; s2 += v * v;
  }
  __shared__ float sh[256], sh2[256];
  sh[threadIdx.x] = s; sh2[threadIdx.x] = s2;
  __syncthreads();
  for (int st = 128; st > 0; st >>= 1) {
    if (threadIdx.x < st) {
      sh[threadIdx.x] += sh[threadIdx.x + st];
      sh2[threadIdx.x] += sh2[threadIdx.x + st];
    }
    __syncthreads();
  }
  if (threadIdx.x == 0) {
    float m = sh[0] / (float)SEQ_N;
    float var = sh2[0] / (float)SEQ_N - m * m;
    float sd = sqrtf(var + 1e-5f);
    mean[bc] = m; stdev[bc] = sd; rstd[bc] = 1.f / sd;
  }
}

// ---------------------------------------------------------------------------
// 2) fused normalize + delay-embed + linear SSM scan.
//    Output M0 in (bc, t, i) layout, i = d*4 + k. float4 coalesced stores.
// ---------------------------------------------------------------------------
__global__ void scan_kernel(const float* __restrict__ x, const float* __restrict__ mean,
                            const float* __restrict__ rstd, const float* __restrict__ Amat,
                            const float* __restrict__ Bv, float* __restrict__ M0) {
  const int tid = threadIdx.x;
  const int d = tid & 31;
  const int bc = blockIdx.x * 8 + (tid >> 5);
  const int b = bc >> 6, c = bc & 63;
  const int j = d & 7, p = d >> 3;
  float Ar[4][4];
#pragma unroll
  for (int r = 0; r < 4; r++)
#pragma unroll
    for (int q = 0; q < 4; q++) Ar[r][q] = Amat[r * 4 + q];
  float Bb[4];
#pragma unroll
  for (int r = 0; r < 4; r++) Bb[r] = Bv[r];
  const float mu = mean[bc], rs = rstd[bc];
  const float* xp = x + ((size_t)b * SEQ_N + j + p) * C_N + c;
  float* mp = M0 + (size_t)bc * NENC * CK_N + d * 4;
  float s0 = 0.f, s1 = 0.f, s2 = 0.f, s3 = 0.f;
  for (int t = 0; t < NENC; t++) {
    float f = (xp[(size_t)(8 * t) * C_N] - mu) * rs;
    // cst = cst @ A.T + f * Bvec   =>  new[k] = sum_j s[j]*A[k][j] + f*B[k]
    float n0 = s0 * Ar[0][0] + s1 * Ar[0][1] + s2 * Ar[0][2] + s3 * Ar[0][3] + f * Bb[0];
    float n1 = s0 * Ar[1][0] + s1 * Ar[1][1] + s2 * Ar[1][2] + s3 * Ar[1][3] + f * Bb[1];
    float n2 = s0 * Ar[2][0] + s1 * Ar[2][1] + s2 * Ar[2][2] + s3 * Ar[2][3] + f * Bb[2];
    float n3 = s0 * Ar[3][0] + s1 * Ar[3][1] + s2 * Ar[3][2] + s3 * Ar[3][3] + f * Bb[3];
    s0 = n0; s1 = n1; s2 = n2; s3 = n3;
    float4 v4 = make_float4(s0, s1, s2, s3);
    *(float4*)(mp + (size_t)t * CK_N) = v4;
  }
}

// ---------------------------------------------------------------------------
// 3) one downsample level: v8 = [m[2t], m[2t+1]] per (bc,d);
//    outc = v8 @ W_down_c, outs = v8 @ W_down_s. Outputs in (bc, i, t) layout.
//    Input strides parameterized so level 0 can read the scan's (bc,t,i).
// ---------------------------------------------------------------------------
__global__ void down_kernel(const float* __restrict__ in, float* __restrict__ outc,
                            float* __restrict__ outs, int n2, int sI, int sT,
                            const float* __restrict__ Wdc, const float* __restrict__ Wds) {
  __shared__ float swc[32], sws[32];
  if (threadIdx.x < 32) { swc[threadIdx.x] = Wdc[threadIdx.x]; sws[threadIdx.x] = Wds[threadIdx.x]; }
  __syncthreads();
  const int tt = blockIdx.x * 64 + threadIdx.x;
  if (tt >= n2) return;
  const int d = blockIdx.y;
  const int bc = blockIdx.z;
  const float* bp = in + (size_t)bc * CK_N * (2 * n2);
  float v[8];
  const size_t t0 = (size_t)(2 * tt) * sT;
#pragma unroll
  for (int q = 0; q < 4; q++) {
    v[q]     = bp[(size_t)(d * 4 + q) * sI + t0];
    v[4 + q] = bp[(size_t)(d * 4 + q) * sI + t0 + sT];
  }
  float* oc = outc + (size_t)bc * CK_N * n2 + (size_t)(d * 4) * n2 + tt;
  float* os = outs + (size_t)bc * CK_N * n2 + (size_t)(d * 4) * n2 + tt;
#pragma unroll
  for (int k = 0; k < 4; k++) {
    float ac = 0.f, as = 0.f;
#pragma unroll
    for (int q = 0; q < 8; q++) { ac += v[q] * swc[q * 4 + k]; as += v[q] * sws[q * 4 + k]; }
    oc[(size_t)k * n2] = ac;
    os[(size_t)k * n2] = as;
  }
}

// ---------------------------------------------------------------------------
// 4) build forward DFT matrix F (n x 2*lp): cols [0,lp)=cos, [lp,2lp)=-sin,
//    zero for padded modes x >= l. Exact integer phase reduction.
// ---------------------------------------------------------------------------
__global__ void fill_F(float* __restrict__ Fm, int n, int l, int lp) {
  const int N = 2 * lp;
  const int idx = blockIdx.x * blockDim.x + threadIdx.x;
  if (idx >= n * N) return;
  const int t = idx / N, xx = idx % N;
  float v = 0.f;
  if (xx < lp) {
    if (xx < l) { int ph = (t * xx) % n; v = cosf(6.28318530718f * (float)ph / (float)n); }
  } else {
    int xm = xx - lp;
    if (xm < l) { int ph = (t * xm) % n; v = -sinf(6.28318530718f * (float)ph / (float)n); }
  }
  Fm[idx] = v;
}

// 5) transpose Kw (CK,CK,MODES) -> per-mode dense planes Wt[x][i][o]
__global__ void wt_kernel(const float* __restrict__ Kre, const float* __restrict__ Kim,
                          float* __restrict__ Wre, float* __restrict__ Wim) {
  const int idx = blockIdx.x * blockDim.x + threadIdx.x;
  if (idx >= 32 * CK_N * CK_N) return;
  const int xx = idx >> 14;       // mode
  const int rem = idx & 16383;    // i*128 + o
  Wre[idx] = Kre[(size_t)rem * 32 + xx];
  Wim[idx] = Kim[(size_t)rem * 32 + xx];
}

// ---------------------------------------------------------------------------
// 6) forward DFT as WMMA GEMM: XF[col][r] = sum_t A[r,t] * F[t,col]
//    A in (bc,i,t) layout => affine (lda = n). Output plane-per-column.
//    One wave = one 16-row tile x NT column tiles. Block = 4 waves.
// ---------------------------------------------------------------------------
template <int NT>
__global__ void __launch_bounds__(128) dft_fwd(const float* __restrict__ A,
                                               const float* __restrict__ Fm,
                                               float* __restrict__ XF, int n, int N) {
  const int lane = threadIdx.x & 31;
  const int wid = threadIdx.x >> 5;
  const int m0 = (blockIdx.x * 4 + wid) * 16;
  const int mr = lane & 15, kh = lane >> 4;
  v8f acc[NT];
#pragma unroll
  for (int i = 0; i < NT; i++) acc[i] = v8zero();
  const float* arow = A + (size_t)(m0 + mr) * n;
  for (int k0 = 0; k0 < n; k0 += 4) {
    __builtin_prefetch(arow + k0 + 16, 0, 1);   // global_prefetch_b8
    v2f a; a.x = arow[k0 + 2 * kh]; a.y = arow[k0 + 2 * kh + 1];
#pragma unroll
    for (int nt = 0; nt < NT; nt++) {
      const int col = nt * 16 + mr;
      v2f bf;
      bf.x = Fm[(k0 + 2 * kh) * N + col];
      bf.y = Fm[(k0 + 2 * kh + 1) * N + col];
      acc[nt] = wmma_f32_16x16x4(a, bf, acc[nt]);
    }
  }
#pragma unroll
  for (int nt = 0; nt < NT; nt++) {
    const int col = nt * 16 + mr;
    float* cp = XF + (size_t)col * MROWS + m0 + 8 * kh;
#pragma unroll
    for (int v = 0; v < 8; v++) cp[v] = acc[nt][v];
  }
}

// ---------------------------------------------------------------------------
// 7) per-mode complex GEMM (2048 x 128 x 128) via 4 WMMA chains x 2 N-tiles:
//    OUT_re = XFre*Wre - XFim*Wim ; OUT_im = XFre*Wim + XFim*Wre
// ---------------------------------------------------------------------------
__global__ void __launch_bounds__(128) mode_gemm(const float* __restrict__ XF,
                                                 const float* __restrict__ Wre,
                                                 const float* __restrict__ Wim,
                                                 float* __restrict__ OUTB, int l, int lp) {
  const int lane = threadIdx.x & 31;
  const int wid = threadIdx.x >> 5;
  const int xx = blockIdx.z;
  const int m0 = (blockIdx.x * 4 + wid) * 16;   // bc-tile base
  const int o0 = blockIdx.y * 32;               // two 16-wide o tiles
  const int mr = lane & 15, kh = lane >> 4;
  float* outRe = OUTB + (size_t)xx * MROWS;
  float* outIm = OUTB + (size_t)(lp + xx) * MROWS;
  if (xx >= l) {  // padded mode planes: write zeros (K for the inverse stage)
#pragma unroll
    for (int nt = 0; nt < 2; nt++) {
      const int o = o0 + nt * 16 + mr;
      float* pr = outRe + (size_t)(m0 + 8 * kh) * CK_N + o;
      float* pi = outIm + (size_t)(m0 + 8 * kh) * CK_N + o;
#pragma unroll
      for (int v = 0; v < 8; v++) { pr[(size_t)v * CK_N] = 0.f; pi[(size_t)v * CK_N] = 0.f; }
    }
    return;
  }
  const float* Are = XF + (size_t)xx * MROWS + (size_t)(m0 + mr) * CK_N;
  const float* Aim = XF + (size_t)(lp + xx) * MROWS + (size_t)(m0 + mr) * CK_N;
  const float* Br = Wre + (size_t)xx * (CK_N * CK_N);
  const float* Bi = Wim + (size_t)xx * (CK_N * CK_N);
  v8f rr0 = v8zero(), rr1 = v8zero(), ii0 = v8zero(), ii1 = v8zero();
  v8f ri0 = v8zero(), ri1 = v8zero(), ir0 = v8zero(), ir1 = v8zero();
  for (int k0 = 0; k0 < CK_N; k0 += 4) {
    v2f ar, ai;
    ar.x = Are[k0 + 2 * kh]; ar.y = Are[k0 + 2 * kh + 1];
    ai.x = Aim[k0 + 2 * kh]; ai.y = Aim[k0 + 2 * kh + 1];
    const int r0 = (k0 + 2 * kh) * CK_N, r1 = r0 + CK_N;
    const int c0 = o0 + mr, c1 = o0 + 16 + mr;
    v2f br0, br1, bi0, bi1;
    br0.x = Br[r0 + c0]; br0.y = Br[r1 + c0];
    br1.x = Br[r0 + c1]; br1.y = Br[r1 + c1];
    bi0.x = Bi[r0 + c0]; bi0.y = Bi[r1 + c0];
    bi1.x = Bi[r0 + c1]; bi1.y = Bi[r1 + c1];
    rr0 = wmma_f32_16x16x4(ar, br0, rr0);
    ii0 = wmma_f32_16x16x4(ai, bi0, ii0);
    ri0 = wmma_f32_16x16x4(ar, bi0, ri0);
    ir0 = wmma_f32_16x16x4(ai, br0, ir0);
    rr1 = wmma_f32_16x16x4(ar, br1, rr1);
    ii1 = wmma_f32_16x16x4(ai, bi1, ii1);
    ri1 = wmma_f32_16x16x4(ar, bi1, ri1);
    ir1 = wmma_f32_16x16x4(ai, br1, ir1);
  }
  v8f re0 = rr0 - ii0, im0 = ri0 + ir0;
  v8f re1 = rr1 - ii1, im1 = ri1 + ir1;
  const int o_0 = o0 + mr, o_1 = o0 + 16 + mr;
  float* pr0 = outRe + (size_t)(m0 + 8 * kh) * CK_N + o_0;
  float* pi0 = outIm + (size_t)(m0 + 8 * kh) * CK_N + o_0;
  float* pr1 = outRe + (size_t)(m0 + 8 * kh) * CK_N + o_1;
  float* pi1 = outIm + (size_t)(m0 + 8 * kh) * CK_N + o_1;
#pragma unroll
  for (int v = 0; v < 8; v++) {
    pr0[(size_t)v * CK_N] = re0[v]; pi0[(size_t)v * CK_N] = im0[v];
    pr1[(size_t)v * CK_N] = re1[v]; pi1[(size_t)v * CK_N] = im1[v];
  }
}

// ---------------------------------------------------------------------------
// 8) single-column inverse rFFT: y[r] = sum_x w_x/n * (Re*cos - Im*sin)
// ---------------------------------------------------------------------------
__global__ void idft_col(const float* __restrict__ OUTB, float* __restrict__ y,
                         int l, int lp, int n, int tstar) {
  const int r = blockIdx.x * blockDim.x + threadIdx.x;
  if (r >= MROWS) return;
  const float inv_n = 1.f / (float)n;
  float acc = 0.f;
  for (int xx = 0; xx < l; xx++) {
    const int ph = (tstar * xx) % n;
    const float th = 6.28318530718f * (float)ph / (float)n;
    const float w = (xx == 0 || 2 * xx == n) ? inv_n : 2.f * inv_n;
    acc += OUTB[(size_t)xx * MROWS + r] * (w * cosf(th))
         - OUTB[(size_t)(lp + xx) * MROWS + r] * (w * sinf(th));
  }
  y[r] = acc;
}

// ---------------------------------------------------------------------------
// 9) collapsed upsample chain: three 8->4 contractions with W_up_1 only
//    (all selected time indices are odd => W_up_0 is dead).
// ---------------------------------------------------------------------------
__global__ void final_m63(const float* __restrict__ y_mc, const float* __restrict__ y_ms2,
                          const float* __restrict__ y_ms1, const float* __restrict__ y_ms0,
                          const float* __restrict__ Wu1, float* __restrict__ m63) {
  const int idx = blockIdx.x * blockDim.x + threadIdx.x;
  if (idx >= BC_N * D_N) return;
  const int bc = idx >> 5, d = idx & 31;
  const int base = bc * CK_N + d * 4;
  float W[32];
#pragma unroll
  for (int q = 0; q < 32; q++) W[q] = Wu1[q];
  float t2[4], t1[4], t0[4];
#pragma unroll
  for (int k = 0; k < 4; k++) {
    float a = 0.f;
#pragma unroll
    for (int q = 0; q < 4; q++)
      a += y_mc[base + q] * W[q * 4 + k] + y_ms2[base + q] * W[(q + 4) * 4 + k];
    t2[k] = a;
  }
#pragma unroll
  for (int k = 0; k < 4; k++) {
    float a = 0.f;
#pragma unroll
    for (int q = 0; q < 4; q++)
      a += t2[q] * W[q * 4 + k] + y_ms1[base + q] * W[(q + 4) * 4 + k];
    t1[k] = a;
  }
#pragma unroll
  for (int k = 0; k < 4; k++) {
    float a = 0.f;
#pragma unroll
    for (int q = 0; q < 4; q++)
      a += t1[q] * W[q * 4 + k] + y_ms0[base + q] * W[(q + 4) * 4 + k];
    t0[k] = a;
  }
#pragma unroll
  for (int k = 0; k < 4; k++) m63[(size_t)idx * 4 + k] = t0[k];
}

// ---------------------------------------------------------------------------
// 10) Cmat projection + inverse delay-embedding + de-normalization
// ---------------------------------------------------------------------------
__global__ void write_out(const float* __restrict__ m63, const float* __restrict__ Cmat,
                          const float* __restrict__ mean, const float* __restrict__ stdev,
                          float* __restrict__ out) {
  const int total = B_N * PRED_N * C_N;
  const int idx = blockIdx.x * blockDim.x + threadIdx.x;
  if (idx >= total) return;
  const int c = idx & 63;
  const int t = (idx >> 6) % PRED_N;
  const int b = idx / (PRED_N * C_N);
  const int bc = b * C_N + c;
  const int i = t < 3 ? t : 3;             // last-writer wins (DELAY=1)
  const int s = t - i;
  const int td = s >> 3, jj = s & 7;
  const int dd = i * 8 + jj;
  const float* mp = m63 + ((size_t)bc * D_N + dd) * 4;
  float v = mp[0] * Cmat[0 * 64 + td] + mp[1] * Cmat[1 * 64 + td] +
            mp[2] * Cmat[2 * 64 + td] + mp[3] * Cmat[3 * 64 + td];
  out[idx] = v * stdev[bc] + mean[bc];
}

// ---------------------------------------------------------------------------
// host-side spectral driver: forward DFT GEMM -> mode GEMM -> 1-column iDFT
// ---------------------------------------------------------------------------
static void run_spectral(const float* Xin, int n, int l, int lp, int tstar,
                         const float* Fm, float* XF, float* OUTB,
                         const float* Wre, const float* Wim, float* y,
                         hipStream_t stream) {
  const int N = 2 * lp;
  dim3 g1(MROWS / 16 / 4, 1, 1);
  if (N == 64)
    dft_fwd<4><<<g1, 128, 0, stream>>>(Xin, Fm, XF, n, N);
  else
    dft_fwd<3><<<g1, 128, 0, stream>>>(Xin, Fm, XF, n, N);
  dim3 g2(BC_N / 16 / 4, CK_N / 32, lp);
  mode_gemm<<<g2, 128, 0, stream>>>(XF, Wre, Wim, OUTB, l, lp);
  idft_col<<<(MROWS + 255) / 256, 256, 0, stream>>>(OUTB, y, l, lp, n, tstar);
}

extern "C" void kernel_launch(void* const* d_in, const int* in_sizes, int n_in,
                              void* d_out, int out_size, void* d_ws, size_t ws_size,
                              hipStream_t stream) {
  const float* x_enc = (const float*)d_in[0];
  const float* Amat  = (const float*)d_in[4];
  const float* Bvec  = (const float*)d_in[5];
  const float* Cmat  = (const float*)d_in[6];
  const float* Wdc   = (const float*)d_in[7];
  const float* Wds   = (const float*)d_in[8];
  const float* Wu1   = (const float*)d_in[10];
  const float* Kre   = (const float*)d_in[11];
  const float* Kim   = (const float*)d_in[12];
  float* out = (float*)d_out;
  float* ws = (float*)d_ws;

  // ---- workspace layout (floats); XF/OUT overlay the dead scan buffer ----
  float* mean  = ws;
  float* stdev = mean + 2048;
  float* rstd  = stdev + 2048;
  float* F128  = rstd + 2048;              // 128*64
  float* F64   = F128 + 128 * 64;          // 64*64
  float* F32   = F64 + 64 * 64;            // 32*48
  float* Wre   = F32 + 32 * 48;            // 32*128*128
  float* Wim   = Wre + 32 * CK_N * CK_N;
  float* y_ms0 = Wim + 32 * CK_N * CK_N;   // 4 x MROWS single-column results
  float* y_ms1 = y_ms0 + MROWS;
  float* y_ms2 = y_ms1 + MROWS;
  float* y_mc  = y_ms2 + MROWS;
  float* m63   = y_mc + MROWS;             // MROWS
  float* M0    = m63 + MROWS;              // (bc,t,i) n=256 : 67.1M floats
  float* XF    = M0;                       // overlay (M0 dead after level-0)
  float* OUTB  = M0 + (size_t)64 * MROWS;
  float* mc1   = M0 + (size_t)BC_N * NENC * CK_N;       // (bc,i,128)
  float* ms1T  = mc1 + (size_t)BC_N * CK_N * 128;       // (bc,i,128)
  float* mc2   = ms1T;                                  // overlay after spectral(ms1T)
  float* ms2T  = ms1T + (size_t)BC_N * CK_N * 64;
  float* mc3T  = mc1;                                   // overlay after level-1
  float* ms3T  = mc1 + (size_t)BC_N * CK_N * 32;

  stats_kernel<<<BC_N, 256, 0, stream>>>(x_enc, mean, stdev, rstd);
  scan_kernel<<<BC_N / 8, 256, 0, stream>>>(x_enc, mean, rstd, Amat, Bvec, M0);

  fill_F<<<(128 * 64 + 255) / 256, 256, 0, stream>>>(F128, 128, 32, 32);
  fill_F<<<(64 * 64 + 255) / 256, 256, 0, stream>>>(F64, 64, 32, 32);
  fill_F<<<(32 * 48 + 255) / 256, 256, 0, stream>>>(F32, 32, 17, 24);
  wt_kernel<<<(32 * CK_N * CK_N + 255) / 256, 256, 0, stream>>>(Kre, Kim, Wre, Wim);

  // level 0: M0 (bc,t,i) n=256 -> mc1, ms1T (bc,i,128)
  down_kernel<<<dim3(2, 32, BC_N), 64, 0, stream>>>(M0, mc1, ms1T, 128, 1, CK_N, Wdc, Wds);
  run_spectral(ms1T, 128, 32, 32, 31, F128, XF, OUTB, Wre, Wim, y_ms0, stream);

  // level 1: mc1 (bc,i,128) -> mc2, ms2T (bc,i,64)
  down_kernel<<<dim3(1, 32, BC_N), 64, 0, stream>>>(mc1, mc2, ms2T, 64, 128, 1, Wdc, Wds);
  run_spectral(ms2T, 64, 32, 32, 15, F64, XF, OUTB, Wre, Wim, y_ms1, stream);

  // level 2: mc2 (bc,i,64) -> mc3T, ms3T (bc,i,32)
  down_kernel<<<dim3(1, 32, BC_N), 64, 0, stream>>>(mc2, mc3T, ms3T, 32, 64, 1, Wdc, Wds);
  run_spectral(ms3T, 32, 17, 24, 7, F32, XF, OUTB, Wre, Wim, y_ms2, stream);
  run_spectral(mc3T, 32, 17, 24, 7, F32, XF, OUTB, Wre, Wim, y_mc, stream);

  final_m63<<<(BC_N * D_N + 255) / 256, 256, 0, stream>>>(y_mc, y_ms2, y_ms1, y_ms0, Wu1, m63);
  const int total = B_N * PRED_N * C_N;
  write_out<<<(total + 255) / 256, 256, 0, stream>>>(m63, Cmat, mean, stdev, out);
}